// HausdorffDTLoss_17952963297871
// MI455X (gfx1250) — compile-verified
//
#include <hip/hip_runtime.h>
#include <stdint.h>

typedef float        v2f  __attribute__((ext_vector_type(2)));
typedef float        v8f  __attribute__((ext_vector_type(8)));
typedef unsigned int u32x4 __attribute__((ext_vector_type(4)));
typedef int          i32x4 __attribute__((ext_vector_type(4)));
typedef int          i32x8 __attribute__((ext_vector_type(8)));

#ifndef __has_builtin
#define __has_builtin(x) 0
#endif

#if __has_builtin(__builtin_amdgcn_tensor_load_to_lds) && \
    __has_builtin(__builtin_amdgcn_tensor_store_from_lds) && \
    __has_builtin(__builtin_amdgcn_s_wait_tensorcnt)
#define USE_TDM 1
#else
#define USE_TDM 0
#endif

#if __has_builtin(__builtin_amdgcn_wmma_f32_16x16x4_f32)
#define USE_WMMA 1
#else
#define USE_WMMA 0
#endif

#define BIGF  1.0e10f
#define HH    256
#define WW    256
#define NB    4
#define IMG   (HH * WW)      // 65536 px per image
#define FIELD (NB * IMG)     // 262144 px per field buffer

// ---------------------------------------------------------------------------
// TDM helper: issue one 2-D tile DMA (load global->LDS or store LDS->global).
// D# layout per CDNA5 ISA §8.3-8.6.  data_size = 4 bytes.  Wave-uniform args.
// clang-23 toolchain: 6-arg builtin (g0, g1, g2, g3, extra v8i, cpol).
// ---------------------------------------------------------------------------
#if USE_TDM
__device__ __forceinline__ void tdm_tile(unsigned lds_addr, unsigned long long gaddr,
                                         unsigned tile0, unsigned tile1,
                                         unsigned td0, unsigned td1,
                                         unsigned stride0, bool store) {
  u32x4 g0;
  g0[0] = 1u;                                                   // count=1, user mode
  g0[1] = lds_addr;                                             // LDS byte address
  g0[2] = (unsigned)gaddr;                                      // global_addr[31:0]
  g0[3] = ((unsigned)(gaddr >> 32) & 0x01ffffffu) | 0x80000000u; // [56:32] | type=2
  i32x8 g1;
  g1[0] = (int)(2u << 16);                                      // data_size=4B; no mask/pad
  g1[1] = (int)((td0 & 0xffffu) << 16);                         // tensor_dim0[15:0]
  g1[2] = (int)(((td0 >> 16) & 0xffffu) | ((td1 & 0xffffu) << 16)); // td0 hi | td1 lo
  g1[3] = (int)(((td1 >> 16) & 0xffffu) | (tile0 << 16));       // td1 hi | tile_dim0
  g1[4] = (int)(tile1 & 0xffffu);                               // tile_dim1; tile_dim2=0
  g1[5] = (int)stride0;                                         // tensor_dim0_stride lo
  g1[6] = 0;                                                    // stride0 hi | stride1 lo
  g1[7] = 0;                                                    // stride1 hi
  i32x4 g2; g2[0] = 1; g2[1] = 1; g2[2] = 0; g2[3] = 0;         // td2=1, td3=1, tile3=0
  i32x4 g3; g3[0] = 0; g3[1] = (1 << 16); g3[2] = 0; g3[3] = 0; // td4=1, tile4=0
  i32x8 gx = {};                                                // uncharacterized extra group
  if (store) __builtin_amdgcn_tensor_store_from_lds(g0, g1, g2, g3, gx, 0);
  else       __builtin_amdgcn_tensor_load_to_lds(g0, g1, g2, g3, gx, 0);
}
__device__ __forceinline__ unsigned lds_off(const void* p) {
  return (unsigned)(uintptr_t)p;   // low 32 bits of generic ptr = LDS byte address
}
#endif

// ---------------------------------------------------------------------------
// Kernel 1: per-image "any foreground" flags (pred -> flags[0..3], tgt -> [4..7])
// ---------------------------------------------------------------------------
__global__ void k_fgany(const float* __restrict__ pred, const float* __restrict__ tgt,
                        float* __restrict__ flags) {
  const int b = blockIdx.x, src = blockIdx.y;
  const float* img = (src == 0 ? pred : tgt) + (size_t)b * IMG;
  int any = 0;
  for (int i = threadIdx.x; i < IMG; i += 256) any |= (img[i] > 0.5f) ? 1 : 0;
  __shared__ int s[256];
  s[threadIdx.x] = any;
  __syncthreads();
  for (int o = 128; o > 0; o >>= 1) {
    if ((int)threadIdx.x < o) s[threadIdx.x] |= s[threadIdx.x + o];
    __syncthreads();
  }
  if (threadIdx.x == 0) flags[src * 4 + b] = s[0] ? 1.0f : 0.0f;
}

// ---------------------------------------------------------------------------
// Kernel 2: min-plus pass along W.  One block per (row y, image b, field f).
// f: 0=pred-fg 1=pred-bg 2=tgt-fg 3=tgt-bg.  Row staged in LDS via TDM.
// ---------------------------------------------------------------------------
__global__ void k_rowpass(const float* __restrict__ pred, const float* __restrict__ tgt,
                          float* __restrict__ tmp) {
  const int y = blockIdx.x, b = blockIdx.y, f = blockIdx.z;
  const float* src = (f < 2 ? pred : tgt);
  const int bg = f & 1;
  const float* row = src + (size_t)b * IMG + (size_t)y * WW;
  __shared__ float simg[WW];
  __shared__ float sf[WW];
#if USE_TDM
  if (threadIdx.x < 32) {   // wave 0 issues the DMA; descriptors are wave-uniform
    tdm_tile(lds_off(simg), (unsigned long long)(uintptr_t)row,
             /*tile*/ WW, 1, /*tensor*/ WW, 1, /*stride0*/ WW, /*store*/ false);
    __builtin_amdgcn_s_wait_tensorcnt(0);
  }
  __syncthreads();
#else
  simg[threadIdx.x] = row[threadIdx.x];
  __syncthreads();
#endif
  const int t = threadIdx.x;
  int m = simg[t] > 0.5f ? 1 : 0;
  if (bg) m ^= 1;
  sf[t] = m ? BIGF : 0.0f;
  __syncthreads();
  // out[t] = min_j sf[j] + (t-j)^2   ((t-j)^2 <= 65025 is exact in f32)
  float best = 3.0e38f;
  float d = (float)t;
  #pragma unroll 8
  for (int j = 0; j < WW; ++j) {
    best = fminf(best, __builtin_fmaf(d, d, sf[j]));
    d -= 1.0f;
  }
  tmp[(size_t)f * FIELD + (size_t)b * IMG + (size_t)y * WW + t] = best;  // coalesced
}

// ---------------------------------------------------------------------------
// Kernel 3: min-plus pass along H, in place on tmp; clamp to BIG (squared DT).
// Strided column gathered/scattered by the Tensor Data Mover.
// ---------------------------------------------------------------------------
__global__ void k_colpass(float* __restrict__ tmp) {
  const int x = blockIdx.x, b = blockIdx.y, f = blockIdx.z;
  float* col = tmp + (size_t)f * FIELD + (size_t)b * IMG + x;
  __shared__ float sin_[HH];
  __shared__ float sout[HH];
#if USE_TDM
  if (threadIdx.x < 32) {
    tdm_tile(lds_off(sin_), (unsigned long long)(uintptr_t)col,
             /*tile*/ 1, HH, /*tensor*/ WW, HH, /*stride0*/ WW, /*store*/ false);
    __builtin_amdgcn_s_wait_tensorcnt(0);
  }
  __syncthreads();
#else
  sin_[threadIdx.x] = col[(size_t)threadIdx.x * WW];
  __syncthreads();
#endif
  const int t = threadIdx.x;
  float best = 3.0e38f;
  float d = (float)t;
  #pragma unroll 8
  for (int j = 0; j < HH; ++j) {
    best = fminf(best, __builtin_fmaf(d, d, sin_[j]));
    d -= 1.0f;
  }
  sout[t] = fminf(best, BIGF);   // jnp.minimum(f, BIG) -> squared DT
  __syncthreads();
#if USE_TDM
  if (threadIdx.x < 32) {
    tdm_tile(lds_off(sout), (unsigned long long)(uintptr_t)col,
             /*tile*/ 1, HH, /*tensor*/ WW, HH, /*stride0*/ WW, /*store*/ true);
    __builtin_amdgcn_s_wait_tensorcnt(0);
  }
#else
  col[(size_t)t * WW] = sout[t];
#endif
}

// ---------------------------------------------------------------------------
// Kernel 4: per-pixel loss term + deterministic block reduction.
// Block sum via chained v_wmma_f32_16x16x4_f32 with B = ones (exact f32).
// ---------------------------------------------------------------------------
__global__ void k_loss(const float* __restrict__ pred, const float* __restrict__ tgt,
                       const float* __restrict__ tmp, const float* __restrict__ flags,
                       float* __restrict__ partials) {
  const int p = blockIdx.x * 256 + threadIdx.x;   // 1024 blocks cover 262144 px exactly
  const int b = p >> 16;
  const float q0 = tmp[p];                         // pred fg, squared & clamped
  const float q1 = tmp[FIELD + p];                 // pred bg
  const float q2 = tmp[2 * FIELD + p];             // tgt fg
  const float q3 = tmp[3 * FIELD + p];             // tgt bg
  const float pf = (flags[b]     > 0.5f) ? (sqrtf(q0) + sqrtf(q1)) : 0.0f;
  const float tf = (flags[4 + b] > 0.5f) ? (sqrtf(q2) + sqrtf(q3)) : 0.0f;
  const float e = pred[p] - tgt[p];
  const float term = e * e * (pf * pf + tf * tf);  // ALPHA = 2
  __shared__ float sterm[256];
  sterm[threadIdx.x] = term;
  __syncthreads();
#if USE_WMMA
  if (threadIdx.x < 32) {                          // full wave32, EXEC all ones
    v2f ones; ones[0] = 1.0f; ones[1] = 1.0f;
    v8f c = {};
    #pragma unroll
    for (int it = 0; it < 4; ++it) {               // 4 x 64 values
      v2f a;
      a[0] = sterm[it * 64 + 2 * threadIdx.x];
      a[1] = sterm[it * 64 + 2 * threadIdx.x + 1];
      c = __builtin_amdgcn_wmma_f32_16x16x4_f32(false, a, false, ones,
                                                (short)0, c, false, false);
    }
    float s = c[0] + c[1] + c[2] + c[3] + c[4] + c[5] + c[6] + c[7];
    for (int o = 16; o >= 1; o >>= 1) s += __shfl_xor(s, o, 32);
    if (threadIdx.x == 0) partials[blockIdx.x] = s * (1.0f / 16.0f);  // SumD = 16*Sum(a)
  }
#else
  for (int o = 128; o > 0; o >>= 1) {
    if ((int)threadIdx.x < o) sterm[threadIdx.x] += sterm[threadIdx.x + o];
    __syncthreads();
  }
  if (threadIdx.x == 0) partials[blockIdx.x] = sterm[0];
#endif
}

// ---------------------------------------------------------------------------
// Kernel 5: single-wave final reduction of 1024 partials via WMMA; write mean.
// ---------------------------------------------------------------------------
__global__ void k_final(const float* __restrict__ partials, float* __restrict__ out) {
  const int lane = threadIdx.x;
#if USE_WMMA
  v2f ones; ones[0] = 1.0f; ones[1] = 1.0f;
  v8f c = {};
  #pragma unroll
  for (int it = 0; it < 16; ++it) {               // 16 x 64 = 1024 partials
    v2f a;
    a[0] = partials[it * 64 + 2 * lane];
    a[1] = partials[it * 64 + 2 * lane + 1];
    c = __builtin_amdgcn_wmma_f32_16x16x4_f32(false, a, false, ones,
                                              (short)0, c, false, false);
  }
  float s = c[0] + c[1] + c[2] + c[3] + c[4] + c[5] + c[6] + c[7];
  for (int o = 16; o >= 1; o >>= 1) s += __shfl_xor(s, o, 32);
  if (lane == 0) out[0] = s * (1.0f / 16.0f) * (1.0f / (float)(NB * IMG));
#else
  float s = 0.0f;
  for (int i = lane; i < 1024; i += 32) s += partials[i];
  for (int o = 16; o >= 1; o >>= 1) s += __shfl_xor(s, o, 32);
  if (lane == 0) out[0] = s * (1.0f / (float)(NB * IMG));
#endif
}

// ---------------------------------------------------------------------------
extern "C" void kernel_launch(void* const* d_in, const int* in_sizes, int n_in,
                              void* d_out, int out_size, void* d_ws, size_t ws_size,
                              hipStream_t stream) {
  (void)in_sizes; (void)n_in; (void)out_size; (void)ws_size;
  const float* pred = (const float*)d_in[0];
  const float* tgt  = (const float*)d_in[1];
  float* ws       = (float*)d_ws;
  float* tmp      = ws;                       // 4 * FIELD floats (4 MB)
  float* partials = ws + 4 * (size_t)FIELD;   // 1024 floats
  float* flags    = partials + 1024;          // 8 floats
  float* out      = (float*)d_out;

  k_fgany  <<<dim3(NB, 2, 1),      256, 0, stream>>>(pred, tgt, flags);
  k_rowpass<<<dim3(HH, NB, 4),     256, 0, stream>>>(pred, tgt, tmp);
  k_colpass<<<dim3(WW, NB, 4),     256, 0, stream>>>(tmp);
  k_loss   <<<dim3(FIELD / 256),   256, 0, stream>>>(pred, tgt, tmp, flags, partials);
  k_final  <<<dim3(1),              32, 0, stream>>>(partials, out);
}